// determine_9964324127178
// MI455X (gfx1250) — compile-verified
//
#include <hip/hip_runtime.h>

// out[b, j] = p_cls[b, j>>8] * p_subcls[b, j>>6] * p_maingrp[b, j>>5]
//           * p_subgrp[b, j>>3] * p_fi[b, j]
//
// One 256-thread block per row b. Thread t handles leaves j = 8t .. 8t+7,
// whose ancestor chain is exactly (t>>5, t>>3, t>>2, t) -> one scalar per
// thread. Bandwidth-bound streaming kernel:
//   - p_subgrp row staged via gfx1250 async-to-LDS DMA (ASYNCcnt path),
//     overlapped with the upper-level ancestor product and fi loads.
//   - p_fi read / out written with non-temporal b128 ops (use-once streams;
//     keep the 192MB L2 for the small, reused ancestor tables).

typedef float v4f __attribute__((ext_vector_type(4)));  // native clang vector,
                                                        // valid for NT builtins

__global__ __launch_bounds__(256) void hier_prob_product_kernel(
    const float* __restrict__ p_cls,      // [B,    8]
    const float* __restrict__ p_subcls,   // [B,   32]
    const float* __restrict__ p_maingrp,  // [B,   64]
    const float* __restrict__ p_subgrp,   // [B,  256]
    const float* __restrict__ p_fi,       // [B, 2048]
    float* __restrict__ out)              // [B, 2048]
{
    __shared__ float s_subgrp[256];

    const unsigned b = blockIdx.x;
    const unsigned t = threadIdx.x;

#if defined(__gfx1250__)
    // Per-lane async DMA: p_subgrp[b, t] -> s_subgrp[t]. Tracked by ASYNCcnt.
    {
        const unsigned lds_byte = (unsigned)(unsigned long long)(&s_subgrp[t]);
        const unsigned gl_byte  = ((b << 8) + t) << 2;   // < 2^31, GVS mode
        asm volatile("global_load_async_to_lds_b32 %0, %1, %2"
                     :
                     : "v"(lds_byte), "v"(gl_byte), "s"(p_subgrp)
                     : "memory");
    }
#else
    s_subgrp[t] = p_subgrp[b * 256u + t];
#endif

    // Upper three hierarchy levels while the async copy is in flight.
    // Broadcast-friendly loads: 8 + 32 + 64 distinct floats per row, cached.
    const float u = p_cls    [b *  8u + (t >> 5)]
                  * p_subcls [b * 32u + (t >> 3)]
                  * p_maingrp[b * 64u + (t >> 2)];

    const size_t rowoff = (size_t)b * 2048u + (size_t)t * 8u;  // 32B aligned
    const v4f* fi4 = reinterpret_cast<const v4f*>(p_fi + rowoff);
    v4f f0 = __builtin_nontemporal_load(fi4 + 0);   // global_load_b128 NT
    v4f f1 = __builtin_nontemporal_load(fi4 + 1);

#if defined(__gfx1250__)
    // Each thread reads back only the LDS word its own DMA wrote, so draining
    // ASYNCcnt is sufficient -- no workgroup barrier required.
    asm volatile("s_wait_asynccnt 0" ::: "memory");
#else
    __syncthreads();
#endif

    const float a = u * s_subgrp[t];   // full ancestor product for this subgroup

    f0 *= a;
    f1 *= a;

    v4f* o4 = reinterpret_cast<v4f*>(out + rowoff);
    __builtin_nontemporal_store(f0, o4 + 0);        // global_store_b128 NT
    __builtin_nontemporal_store(f1, o4 + 1);
}

extern "C" void kernel_launch(void* const* d_in, const int* in_sizes, int n_in,
                              void* d_out, int out_size, void* d_ws, size_t ws_size,
                              hipStream_t stream) {
    (void)n_in; (void)out_size; (void)d_ws; (void)ws_size;

    const float* p_cls     = (const float*)d_in[0];
    const float* p_subcls  = (const float*)d_in[1];
    const float* p_maingrp = (const float*)d_in[2];
    const float* p_subgrp  = (const float*)d_in[3];
    const float* p_fi      = (const float*)d_in[4];
    float*       out       = (float*)d_out;

    const int B = in_sizes[4] / 2048;   // 16384 rows; one block per row
    hipLaunchKernelGGL(hier_prob_product_kernel,
                       dim3(B), dim3(256), 0, stream,
                       p_cls, p_subcls, p_maingrp, p_subgrp, p_fi, out);
}